// SpectralConv1d_76192719831978
// MI455X (gfx1250) — compile-verified
//
#include <hip/hip_runtime.h>

// ---------------------------------------------------------------------------
// SpectralConv1d (Hartley modes) for MI455X / gfx1250, fp32 WMMA path.
//   stage1: Xht_low = X[4096x8192] * Basis1[8192x32]      (V_WMMA_F32_16X16X4_F32)
//   stage1b: low = (sum of K-split partials) * W          (deterministic reduce)
//   stage2: Out = low[4096x32] * Basis2[32x4098]          (V_WMMA_F32_16X16X4_F32)
// ---------------------------------------------------------------------------

typedef __attribute__((ext_vector_type(2))) float v2f;
typedef __attribute__((ext_vector_type(8))) float v8f;

#define BATCH   64
#define CH      64
#define NN      8192
#define MODES   32
#define ROWS    (BATCH * CH)        // 4096
#define HLEN    16382               // 2*(NN-1), hfft output length
#define OUTLEN  4097                // NN/2 + 1, ihfft input length
#define QLEN    2049                // OUTLEN//2 + 1, complex outputs per row
#define N2      (QLEN * 2)          // 4098 interleaved floats per row
#define N2PAD   4112                // 16 * 257, padded for 16-wide N tiles
#define KSPLIT  4
#define KCHUNK  (NN / KSPLIT)       // 2048

#define TWO_PI 6.28318530717958647692f

// ---- Basis1[j][t] = c_j * cos(2*pi*j*t / 16382), c_0 = c_{N-1} = 1 else 2 ----
__global__ void gen_basis1(float* __restrict__ b1) {
  int idx = blockIdx.x * blockDim.x + threadIdx.x;
  if (idx >= NN * MODES) return;
  int j = idx / MODES;
  int t = idx % MODES;
  float w = (j == 0 || j == NN - 1) ? 1.0f : 2.0f;
  int ph = (j * t) % HLEN;                       // exact integer phase reduction
  float ang = (float)ph * (TWO_PI / (float)HLEN);
  b1[idx] = w * cosf(ang);
}

// ---- Basis2[k][2q+c] = {cos,sin}(2*pi*k*q / 4097) / 4097, padded to N2PAD ----
__global__ void gen_basis2(float* __restrict__ b2) {
  int idx = blockIdx.x * blockDim.x + threadIdx.x;
  if (idx >= MODES * N2PAD) return;
  int k = idx / N2PAD;
  int j = idx % N2PAD;
  float v = 0.0f;
  if (j < N2) {
    int q = j >> 1;
    int ph = (k * q) % OUTLEN;                   // exact integer phase reduction
    float ang = (float)ph * (TWO_PI / (float)OUTLEN);
    float s, c;
    sincosf(ang, &s, &c);
    v = ((j & 1) ? s : c) * (1.0f / (float)OUTLEN);
  }
  b2[idx] = v;
}

// ---- Stage 1: K-split GEMM, one 16x16 tile per wave, fp32 WMMA -------------
__global__ void __launch_bounds__(256)
stage1(const float* __restrict__ x, const float* __restrict__ b1,
       float* __restrict__ partial) {
  const int wave = threadIdx.x >> 5;
  const int lane = threadIdx.x & 31;
  const int task = blockIdx.x * 8 + wave;        // 2048 tasks = 4 ksplits x 512 tiles
  const int ks   = task >> 9;                    // K-split index 0..3
  const int tile = task & 511;                   // 256 M-tiles x 2 N-tiles
  const int rowBase = (tile >> 1) * 16;
  const int colBase = (tile & 1) * 16;
  const int hi  = lane >> 4;                     // half-wave select (K pair)
  const int l15 = lane & 15;

  const float* xrow = x  + (size_t)(rowBase + l15) * NN;
  const float* bcol = b1 + colBase + l15;
  const int k0 = ks * KCHUNK;
  const int k1 = k0 + KCHUNK;

  v8f acc = {};
  for (int kb = k0; kb < k1; kb += 4) {
    const int kk = kb + hi * 2;
    v2f a, b;
    a.x = xrow[kk];
    a.y = xrow[kk + 1];
    b.x = bcol[(size_t)kk * MODES];
    b.y = bcol[(size_t)(kk + 1) * MODES];
    acc = __builtin_amdgcn_wmma_f32_16x16x4_f32(
        false, a, false, b, (short)0, acc, false, false);
  }

  float* p = partial + (size_t)ks * ROWS * MODES;
#pragma unroll
  for (int i = 0; i < 8; ++i) {
    const int row = rowBase + i + hi * 8;        // C/D layout: M = i + 8*(lane>=16)
    const int col = colBase + l15;               // N = lane & 15
    p[(size_t)row * MODES + col] = acc[i];
  }
}

// ---- Stage 1b: deterministic K-split reduce + elementwise weight multiply ---
__global__ void stage1b(const float* __restrict__ partial,
                        const float* __restrict__ w,
                        float* __restrict__ low) {
  int idx = blockIdx.x * blockDim.x + threadIdx.x;
  if (idx >= ROWS * MODES) return;
  float s = 0.0f;
#pragma unroll
  for (int ks = 0; ks < KSPLIT; ++ks)
    s += partial[(size_t)ks * ROWS * MODES + idx];
  low[idx] = s * w[idx];
}

// ---- Stage 2: low[4096x32] * Basis2[32xN2PAD], guarded store ---------------
__global__ void __launch_bounds__(256)
stage2(const float* __restrict__ low, const float* __restrict__ b2,
       float* __restrict__ out) {
  const int wave = threadIdx.x >> 5;
  const int lane = threadIdx.x & 31;
  const int tile = blockIdx.x * 8 + wave;        // 256 * 257 tiles, exact grid
  const int ntN  = N2PAD / 16;                   // 257
  const int rowBase = (tile / ntN) * 16;
  const int colBase = (tile % ntN) * 16;
  const int hi  = lane >> 4;
  const int l15 = lane & 15;

  const float* arow = low + (size_t)(rowBase + l15) * MODES;
  const float* bcol = b2 + colBase + l15;

  v8f acc = {};
#pragma unroll
  for (int kb = 0; kb < MODES; kb += 4) {
    const int kk = kb + hi * 2;
    v2f a, b;
    a.x = arow[kk];
    a.y = arow[kk + 1];
    b.x = bcol[(size_t)kk * N2PAD];
    b.y = bcol[(size_t)(kk + 1) * N2PAD];
    acc = __builtin_amdgcn_wmma_f32_16x16x4_f32(
        false, a, false, b, (short)0, acc, false, false);
  }

#pragma unroll
  for (int i = 0; i < 8; ++i) {
    const int row = rowBase + i + hi * 8;
    const int col = colBase + l15;
    if (col < N2)                                // drop the 14-column pad tail
      out[(size_t)row * N2 + col] = acc[i];
  }
}

// ---------------------------------------------------------------------------
extern "C" void kernel_launch(void* const* d_in, const int* in_sizes, int n_in,
                              void* d_out, int out_size, void* d_ws, size_t ws_size,
                              hipStream_t stream) {
  const float* x = (const float*)d_in[0];        // [64, 64, 8192] fp32
  const float* w = (const float*)d_in[1];        // [64, 64, 32]   fp32
  float* out = (float*)d_out;                    // [64, 64, 2049] complex64 -> interleaved f32

  // Workspace layout (bytes):
  //   b1      : 8192*32*4             = 1,048,576
  //   partial : 4*4096*32*4           = 2,097,152
  //   low     : 4096*32*4             =   524,288
  //   b2      : 32*4112*4             =   526,336   (total ~4.1 MB)
  char* ws = (char*)d_ws;
  float* b1      = (float*)ws;
  float* partial = (float*)(ws + (size_t)NN * MODES * 4);
  float* low     = (float*)(ws + (size_t)NN * MODES * 4 + (size_t)KSPLIT * ROWS * MODES * 4);
  float* b2      = (float*)(ws + (size_t)NN * MODES * 4 + (size_t)KSPLIT * ROWS * MODES * 4
                               + (size_t)ROWS * MODES * 4);

  gen_basis1<<<(NN * MODES + 255) / 256, 256, 0, stream>>>(b1);
  gen_basis2<<<(MODES * N2PAD + 255) / 256, 256, 0, stream>>>(b2);
  stage1  <<<(KSPLIT * 512) / 8, 256, 0, stream>>>(x, b1, partial);   // 256 blocks
  stage1b <<<(ROWS * MODES + 255) / 256, 256, 0, stream>>>(partial, w, low);
  stage2  <<<(256 * (N2PAD / 16)) / 8, 256, 0, stream>>>(low, b2, out); // 8224 blocks
}